// MLMM_electrostatics_70806830842318
// MI455X (gfx1250) — compile-verified
//
#include <hip/hip_runtime.h>
#include <stdint.h>

// MLMM electrostatics for MI455X (gfx1250).
// Bandwidth-bound (~480 MB streamed, ~21 us @ 23.3 TB/s). CDNA5 path:
// double-buffered global_load_async_to_lds_b128/b32 staging of the wide
// per-pair streams (vec, outer) + s_wait_asynccnt, conflict-free LDS reads,
// global_prefetch for the narrow streams, f32 atomics for the scatter.

namespace {

constexpr float KE_F     = 14.399645351950548f;
constexpr float CUTOFF_F = 10.0f;
constexpr int   BLOCK    = 256;   // 8 wave32s
constexpr int   TILE     = 256;   // pairs per block-iteration (== BLOCK)

// ---------- gfx1250 async-copy primitives (inline asm; toolchain-portable) ----
__device__ __forceinline__ uint32_t lds_addr32(const void* p) {
  // addrspace(3) flat address: low 32 bits are the LDS byte offset.
  return (uint32_t)(uintptr_t)p;
}
__device__ __forceinline__ void async_b32(uint32_t lds, const float* g) {
  asm volatile("global_load_async_to_lds_b32 %0, %1, off th:TH_LOAD_NT"
               :: "v"(lds), "v"(g) : "memory");
}
__device__ __forceinline__ void async_b128(uint32_t lds, const float* g) {
  asm volatile("global_load_async_to_lds_b128 %0, %1, off th:TH_LOAD_NT"
               :: "v"(lds), "v"(g) : "memory");
}
__device__ __forceinline__ void wait_async_le6() {   // next tile's 6 may remain
  asm volatile("s_wait_asynccnt 0x6" ::: "memory");
}
__device__ __forceinline__ void wait_async_0() {
  asm volatile("s_wait_asynccnt 0x0" ::: "memory");
}

// Issue exactly 6 async instructions per wave, all lanes active:
//   vec  : 768 dwords  -> 3 x b32
//   outer: 2304 dwords -> 2 x b128 (dwords [0,2048)) + 1 x b32 (dwords [2048,2304))
__device__ __forceinline__ void stage_tile(int tile, int tid,
                                           const float* __restrict__ vec,
                                           const float* __restrict__ outer,
                                           float* sv, float* so) {
  const size_t base = (size_t)tile * TILE;
  const float* gv = vec   + base * 3;   // 3 KB, 16B-aligned region
  const float* go = outer + base * 9;   // 9 KB, 16B-aligned region
  async_b32 (lds_addr32(sv + tid),             gv + tid);
  async_b32 (lds_addr32(sv + tid + 256),       gv + tid + 256);
  async_b32 (lds_addr32(sv + tid + 512),       gv + tid + 512);
  async_b128(lds_addr32(so + 4 * tid),         go + 4 * tid);
  async_b128(lds_addr32(so + 4 * tid + 1024),  go + 4 * tid + 1024);
  async_b32 (lds_addr32(so + 2048 + tid),      go + 2048 + tid);
}

// ---------- shared physics ----------------------------------------------------
__device__ __forceinline__ float pair_e(float d, float vx, float vy, float vz,
                                        const float o[9], int u, int v,
                                        const float* __restrict__ q_ml,
                                        const float* __restrict__ q_mm,
                                        const float* __restrict__ mu,
                                        const float* __restrict__ quad) {
  const float qu = q_ml[u];
  const float qv = q_mm[v];
  const float* __restrict__ mp = mu + 3 * (size_t)u;
  const float dvm = vx * mp[0] + vy * mp[1] + vz * mp[2];

  const float* __restrict__ Qp = quad + 9 * (size_t)u;
  float s = 0.0f, trQ;
  {
    float Q[9];
#pragma unroll
    for (int k = 0; k < 9; ++k) Q[k] = Qp[k];
#pragma unroll
    for (int k = 0; k < 9; ++k) s = fmaf(o[k], Q[k], s);
    trQ = Q[0] + Q[4] + Q[8];
  }
  const float trO = o[0] + o[4] + o[8];
  const float quadterm = s - (trO * (1.0f / 3.0f)) * trQ;

  const float chi  = 1.0f / d;
  const float chi2 = chi * chi;
  const float chi3 = chi2 * chi;
  const float chi5 = chi3 * chi2;
  return KE_F * (qu * qv * chi - qv * dvm * chi3 + qv * quadterm * chi5);
}

// ---------- kernels -----------------------------------------------------------
__global__ __launch_bounds__(BLOCK) void mlmm_init_out(
    const float* __restrict__ atomic_energies, float* __restrict__ out, int n) {
  int i = blockIdx.x * BLOCK + threadIdx.x;
  if (i < n) out[i] = atomic_energies[i];
}

__global__ __launch_bounds__(BLOCK) void mlmm_pair_tiles(
    const float* __restrict__ q_ml, const float* __restrict__ q_mm,
    const float* __restrict__ mu, const float* __restrict__ quad,
    const float* __restrict__ dist, const float* __restrict__ vec,
    const float* __restrict__ outer, const int* __restrict__ idx_u,
    const int* __restrict__ idx_v, float* __restrict__ out, int ntiles) {
  __shared__ __align__(16) float s_vec[2][TILE * 3];
  __shared__ __align__(16) float s_out[2][TILE * 9];

  const int tid  = threadIdx.x;
  const int step = gridDim.x;
  int t = blockIdx.x;
  int b = 0;

  if (t < ntiles) stage_tile(t, tid, vec, outer, s_vec[0], s_out[0]);

  for (; t < ntiles; t += step, b ^= 1) {
    const int nxt = t + step;
    if (nxt < ntiles) {
      stage_tile(nxt, tid, vec, outer, s_vec[b ^ 1], s_out[b ^ 1]);
      // prefetch next tile's narrow streams (global_prefetch_b8)
      const size_t nb = (size_t)nxt * TILE + tid;
      __builtin_prefetch(&dist[nb], 0, 0);
      __builtin_prefetch(&idx_u[nb], 0, 0);
      __builtin_prefetch(&idx_v[nb], 0, 0);
      wait_async_le6();          // current buffer's 6 transfers complete
    } else {
      wait_async_0();
    }
    __syncthreads();             // other waves staged parts of my pairs

    const size_t p = (size_t)t * TILE + tid;
    const float d = __builtin_nontemporal_load(&dist[p]);
    const int   u = __builtin_nontemporal_load(&idx_u[p]);
    const int   v = __builtin_nontemporal_load(&idx_v[p]);

    // conflict-free LDS reads: strides 3 and 9 are coprime with 64 banks
    const float vx = s_vec[b][3 * tid + 0];
    const float vy = s_vec[b][3 * tid + 1];
    const float vz = s_vec[b][3 * tid + 2];
    float o[9];
#pragma unroll
    for (int k = 0; k < 9; ++k) o[k] = s_out[b][9 * tid + k];

    const float e = pair_e(d, vx, vy, vz, o, u, v, q_ml, q_mm, mu, quad);
    if (d <= CUTOFF_F) atomicAdd(&out[u], e);

    __syncthreads();             // buffer b must not be restaged while read
  }
}

__global__ __launch_bounds__(BLOCK) void mlmm_pair_tail(
    const float* __restrict__ q_ml, const float* __restrict__ q_mm,
    const float* __restrict__ mu, const float* __restrict__ quad,
    const float* __restrict__ dist, const float* __restrict__ vec,
    const float* __restrict__ outer, const int* __restrict__ idx_u,
    const int* __restrict__ idx_v, float* __restrict__ out,
    int start, int P) {
  const int p = start + blockIdx.x * BLOCK + threadIdx.x;
  if (p >= P) return;
  const float d = dist[p];
  const int   u = idx_u[p];
  const int   v = idx_v[p];
  const float* vp = vec + 3 * (size_t)p;
  const float* op = outer + 9 * (size_t)p;
  float o[9];
#pragma unroll
  for (int k = 0; k < 9; ++k) o[k] = op[k];
  const float e = pair_e(d, vp[0], vp[1], vp[2], o, u, v, q_ml, q_mm, mu, quad);
  if (d <= CUTOFF_F) atomicAdd(&out[u], e);
}

} // namespace

extern "C" void kernel_launch(void* const* d_in, const int* in_sizes, int n_in,
                              void* d_out, int out_size, void* d_ws, size_t ws_size,
                              hipStream_t stream) {
  const float* q_ml  = (const float*)d_in[0];
  const float* q_mm  = (const float*)d_in[1];
  const float* mu    = (const float*)d_in[2];
  const float* quad  = (const float*)d_in[3];
  const float* e0    = (const float*)d_in[4];
  const float* dist  = (const float*)d_in[5];
  const float* vec   = (const float*)d_in[6];
  const float* outer = (const float*)d_in[7];
  const int*   idx_u = (const int*)d_in[8];
  const int*   idx_v = (const int*)d_in[9];
  float* out = (float*)d_out;

  const int nml = out_size;
  const int P   = in_sizes[5];

  mlmm_init_out<<<(nml + BLOCK - 1) / BLOCK, BLOCK, 0, stream>>>(e0, out, nml);

  const int ntiles = P / TILE;
  const int tail   = P - ntiles * TILE;

  if (ntiles > 0) {
    int blocks = ntiles < 2048 ? ntiles : 2048;
    mlmm_pair_tiles<<<blocks, BLOCK, 0, stream>>>(q_ml, q_mm, mu, quad,
                                                  dist, vec, outer,
                                                  idx_u, idx_v, out, ntiles);
  }
  if (tail > 0) {
    mlmm_pair_tail<<<(tail + BLOCK - 1) / BLOCK, BLOCK, 0, stream>>>(
        q_ml, q_mm, mu, quad, dist, vec, outer, idx_u, idx_v, out,
        ntiles * TILE, P);
  }
}